// GCNEncoder_28673201668465
// MI455X (gfx1250) — compile-verified
//
#include <hip/hip_runtime.h>
#include <math.h>

// ---------------------------------------------------------------------------
// GCNII encoder for MI455X (gfx1250, wave32, WMMA).
//   N=100000 nodes, E=1600000 edges, IN=128, HID=OUT=64, 16 layers.
// GEMMs use v_wmma_f32_16x16x4_f32 (full f32 precision, matches reference).
// Edge aggregation uses float4 gathers + global f32 atomics (the real
// bandwidth bound: ~13 GB of edge traffic; h fits in L2).
// ---------------------------------------------------------------------------

typedef __attribute__((ext_vector_type(2))) float v2f;
typedef __attribute__((ext_vector_type(8))) float v8f;

#define NNODES 100000
#define NEDGES 1600000
#define INC    128
#define HID    64
#define NLAYERS 16

// ---------------------------------------------------------------------------
// WMMA GEMM: out[M,64] = combine( ab_scale * (A[M,K] x B) , res, bias )
//   bt=1: B is [64,K] row-major (torch Linear weight [out,in], used as W^T)
//   bt=0: B is [K,64] row-major (used as plain right-multiply)
// Each wave -> one 16x16 tile. Block = 256 thr = 8 waves = 32 rows x 64 cols.
// ---------------------------------------------------------------------------
__global__ __launch_bounds__(256) void gemm64_wmma(
    const float* __restrict__ A, int K,
    const float* __restrict__ B, int bt,
    const float* __restrict__ bias,     // nullable
    const float* __restrict__ res,      // nullable, [M,64]
    float ab_scale, float res_scale, int do_relu,
    float* __restrict__ out, int M)
{
    const int wave = threadIdx.x >> 5;          // 0..7
    const int lane = threadIdx.x & 31;
    const int m0 = blockIdx.x * 32 + (wave >> 2) * 16;
    const int n0 = (wave & 3) * 16;
    if (m0 >= M) return;

    const int r  = lane & 15;                   // A: row in tile / B: col in tile
    const int kb = (lane >> 4) << 1;            // lanes 0-15 -> K {0,1}, 16-31 -> {2,3}

    v8f acc = {};
    const float* arow = A + (size_t)(m0 + r) * K;

    if (bt) {
        const float* brow = B + (size_t)(n0 + r) * K;   // weight [out,in]
        for (int k = 0; k < K; k += 4) {
            v2f a = *(const v2f*)(arow + k + kb);
            v2f b = *(const v2f*)(brow + k + kb);
            acc = __builtin_amdgcn_wmma_f32_16x16x4_f32(
                false, a, false, b, (short)0, acc, false, false);
        }
    } else {
        for (int k = 0; k < K; k += 4) {
            v2f a = *(const v2f*)(arow + k + kb);
            v2f b;
            b.x = B[(size_t)(k + kb) * 64 + n0 + r];
            b.y = B[(size_t)(k + kb + 1) * 64 + n0 + r];
            acc = __builtin_amdgcn_wmma_f32_16x16x4_f32(
                false, a, false, b, (short)0, acc, false, false);
        }
    }

    // C/D layout: VGPR v -> M = v + 8*(lane>=16), N = lane&15
    const int mofs = (lane >> 4) << 3;
    const int n    = n0 + (lane & 15);
    #pragma unroll
    for (int v = 0; v < 8; ++v) {
        const int m = m0 + v + mofs;
        float val = acc[v] * ab_scale;
        if (res)  val += res_scale * res[(size_t)m * 64 + n];
        if (bias) val += bias[n];
        if (do_relu) val = fmaxf(val, 0.0f);
        out[(size_t)m * 64 + n] = val;
    }
}

// ---------------------------------------------------------------------------
// Edge aggregation: agg[dst] += h[src]. 16 threads/edge, float4 per thread.
// ---------------------------------------------------------------------------
__global__ __launch_bounds__(256) void scatter_add_edges(
    const float* __restrict__ h,
    const long long* __restrict__ src,
    const long long* __restrict__ dst,
    float* __restrict__ agg, int n_edges)
{
    const long long t = (long long)blockIdx.x * blockDim.x + threadIdx.x;
    const long long e = t >> 4;
    if (e >= n_edges) return;
    const int c = ((int)t & 15) << 2;                 // channel chunk 0,4,...,60
    const int s = (int)src[e];
    const int d = (int)dst[e];
    const float4 v = *(const float4*)(h + (size_t)s * HID + c);
    float* p = agg + (size_t)d * HID + c;
    __hip_atomic_fetch_add(p + 0, v.x, __ATOMIC_RELAXED, __HIP_MEMORY_SCOPE_AGENT);
    __hip_atomic_fetch_add(p + 1, v.y, __ATOMIC_RELAXED, __HIP_MEMORY_SCOPE_AGENT);
    __hip_atomic_fetch_add(p + 2, v.z, __ATOMIC_RELAXED, __HIP_MEMORY_SCOPE_AGENT);
    __hip_atomic_fetch_add(p + 3, v.w, __ATOMIC_RELAXED, __HIP_MEMORY_SCOPE_AGENT);
}

__global__ __launch_bounds__(256) void zero_f4(float4* __restrict__ p, int n4)
{
    const int t = blockIdx.x * blockDim.x + threadIdx.x;
    if (t < n4) p[t] = make_float4(0.f, 0.f, 0.f, 0.f);
}

// agg = (1-alpha)*agg + alpha*h0   (in place -> becomes z)
__global__ __launch_bounds__(256) void mix_residual_f4(
    float4* __restrict__ agg, const float4* __restrict__ h0, int n4)
{
    const int t = blockIdx.x * blockDim.x + threadIdx.x;
    if (t >= n4) return;
    const float4 a = agg[t];
    const float4 b = h0[t];
    agg[t] = make_float4(0.9f * a.x + 0.1f * b.x,
                         0.9f * a.y + 0.1f * b.y,
                         0.9f * a.z + 0.1f * b.z,
                         0.9f * a.w + 0.1f * b.w);
}

// ---------------------------------------------------------------------------
extern "C" void kernel_launch(void* const* d_in, const int* in_sizes, int n_in,
                              void* d_out, int out_size, void* d_ws, size_t ws_size,
                              hipStream_t stream)
{
    (void)in_sizes; (void)n_in; (void)out_size; (void)ws_size;
    const float*      x      = (const float*)d_in[0];
    const long long*  ei     = (const long long*)d_in[1];   // int64 [2, E]
    const float*      lin0_w = (const float*)d_in[2];       // [64,128]
    const float*      lin0_b = (const float*)d_in[3];       // [64]
    const float*      lin1_w = (const float*)d_in[4];       // [64,64]
    const float*      lin1_b = (const float*)d_in[5];       // [64]
    const float*      conv_w = (const float*)d_in[6];       // [16,64,64]
    float*            out    = (float*)d_out;               // [N,64]

    const long long* src = ei;
    const long long* dst = ei + NEDGES;

    const size_t feat = (size_t)NNODES * HID;               // 6.4M floats
    float* h0  = (float*)d_ws;
    float* h   = h0 + feat;
    float* agg = h + feat;                                  // doubles as z

    const dim3 blk(256);
    const int gemm_grid  = NNODES / 32;                     // 3125 (exact)
    const int n4         = (int)(feat / 4);                 // 1.6M float4
    const int ew_grid    = (n4 + 255) / 256;
    const int scat_grid  = (int)(((long long)NEDGES * 16 + 255) / 256);

    // h0 = relu(x @ lin0_w^T + lin0_b)
    gemm64_wmma<<<gemm_grid, blk, 0, stream>>>(
        x, INC, lin0_w, /*bt=*/1, lin0_b, nullptr, 1.0f, 0.0f, /*relu=*/1, h0, NNODES);

    const float* cur = h0;
    for (int l = 0; l < NLAYERS; ++l) {
        zero_f4<<<ew_grid, blk, 0, stream>>>((float4*)agg, n4);
        scatter_add_edges<<<scat_grid, blk, 0, stream>>>(cur, src, dst, agg, NEDGES);
        mix_residual_f4<<<ew_grid, blk, 0, stream>>>((float4*)agg, (const float4*)h0, n4);
        const float beta = logf(0.5f / (float)(l + 1) + 1.0f);
        // h = relu((1-beta)*z + beta*(z @ conv_w[l]))
        gemm64_wmma<<<gemm_grid, blk, 0, stream>>>(
            agg, HID, conv_w + (size_t)l * HID * HID, /*bt=*/0,
            nullptr, agg, beta, 1.0f - beta, /*relu=*/1, h, NNODES);
        cur = h;
    }

    // out = h @ lin1_w^T + lin1_b
    gemm64_wmma<<<gemm_grid, blk, 0, stream>>>(
        h, HID, lin1_w, /*bt=*/1, lin1_b, nullptr, 1.0f, 0.0f, /*relu=*/0, out, NNODES);
}